// S4Layer_39548058862111
// MI455X (gfx1250) — compile-verified
//
#include <hip/hip_runtime.h>

typedef __attribute__((ext_vector_type(2))) float v2f;
typedef __attribute__((ext_vector_type(8))) float v8f;

#define T_DIM 4096
#define B_DIM 16
#define D_DIM 128
#define S_DIM 256
#define TG 256                 // timesteps per pass-1 chunk
#define NCH (T_DIM / TG)       // 16 chunks
#define BS (B_DIM * S_DIM)     // 4096

// workspace layout (floats)
#define WS_BBAR 0
#define WS_ABAR (S_DIM * D_DIM)
#define WS_CARR (S_DIM * D_DIM + S_DIM)

__device__ __forceinline__ float clampf_(float v, float lo, float hi) {
    return fminf(fmaxf(v, lo), hi);
}

// ---------------------------------------------------------------------------
// Kernel 0: parameter prep: a_bar[s] and b_bar[s][d] into workspace
// ---------------------------------------------------------------------------
__global__ void s4_prep(const float* __restrict__ log_neg_a,
                        const float* __restrict__ bmat,
                        const float* __restrict__ log_dt,
                        float* __restrict__ ws) {
    int s = blockIdx.x;       // 0..S-1
    int d = threadIdx.x;      // 0..D-1
    float ar = -expf(clampf_(log_neg_a[s], -8.0f, 4.0f));
    float dt = expf(clampf_(log_dt[s], -8.0f, 1.0f));
    float x = dt * ar;
    float ratio = (fabsf(x) < 1e-6f) ? dt : (expm1f(x) / ar);
    ws[WS_BBAR + s * D_DIM + d] = ratio * bmat[s * D_DIM + d];
    if (d == 0) ws[WS_ABAR + s] = expf(x);
}

// ---------------------------------------------------------------------------
// Kernel 1: per-chunk WMMA GEMM + zero-carry local scan, fully in registers.
// One wave (32 lanes) per block: owns (chunk, batch b, 16 states s0..s0+15).
// ---------------------------------------------------------------------------
__global__ void __launch_bounds__(32) s4_pass1(
    const float* __restrict__ inp,   // [T, B, D]
    const float* __restrict__ ws,    // b_bar + a_bar
    float* __restrict__ out,         // [T, B, S]
    float* __restrict__ carr)        // [NCH, B, S] chunk-final local values
{
    const int lane = threadIdx.x;    // 0..31
    const int n    = lane & 15;      // output column within tile (N)
    const int half = lane >> 4;      // 0: t-offsets 0..7, 1: t-offsets 8..15
    const int chunk = blockIdx.x;
    const int b     = blockIdx.y;
    const int s0    = blockIdx.z << 4;
    const int s     = s0 + n;

    const float* bbar = ws + WS_BBAR;
    const float a = ws[WS_ABAR + s];

    // a^(m+1), m = 0..7
    float ap[8];
    ap[0] = a;
#pragma unroll
    for (int m = 1; m < 8; ++m) ap[m] = ap[m - 1] * a;
    const float a8 = ap[7];

    // Preload all B fragments (b_bar is reused for all 16 sub-tiles).
    // B 4x16 tile for k0=4k: VGPR0 = (K=k0+2*half, N=n), VGPR1 = (K=k0+2*half+1, N=n)
    v2f bf[32];
#pragma unroll
    for (int k = 0; k < 32; ++k) {
        bf[k] = *(const v2f*)(bbar + (size_t)s * D_DIM + 4 * k + 2 * half);
    }

    float zc = 0.0f;                   // running local carry (zero at chunk start)
    const int tchunk0 = chunk * TG;

    for (int cc = 0; cc < TG / 16; ++cc) {
        const int tt = tchunk0 + cc * 16;
        // A 16x4 tile: lane holds row M = n (timestep tt+n), K pair selected by half
        const float* arow = inp + ((size_t)(tt + n) * B_DIM + b) * D_DIM + 2 * half;

        v8f acc0 = {};
        v8f acc1 = {};
#pragma unroll
        for (int k = 0; k < 32; k += 2) {
            v2f af0 = *(const v2f*)(arow + 4 * k);
            v2f af1 = *(const v2f*)(arow + 4 * (k + 1));
            acc0 = __builtin_amdgcn_wmma_f32_16x16x4_f32(
                false, af0, false, bf[k],     (short)0, acc0, false, false);
            acc1 = __builtin_amdgcn_wmma_f32_16x16x4_f32(
                false, af1, false, bf[k + 1], (short)0, acc1, false, false);
        }
        v8f u = acc0 + acc1;           // ub tile: u[m] = input proj at t = tt + 8*half + m

        // local scan within this lane's 8 timesteps (zero carry)
#pragma unroll
        for (int m = 1; m < 8; ++m) u[m] = fmaf(a, u[m - 1], u[m]);

        // low half's partial final -> all lanes of the column
        float p7low = __shfl(u[7], n, 32);
        // carry seen by this half: low half uses zc, high half uses low-half final
        float off = (half == 0) ? zc : fmaf(a8, zc, p7low);

        float z = 0.0f;
#pragma unroll
        for (int m = 0; m < 8; ++m) {
            z = fmaf(ap[m], off, u[m]);
            out[((size_t)(tt + 8 * half + m) * B_DIM + b) * S_DIM + s] = z;
        }
        // new carry = z at local t-offset 15 (lives in high half after loop)
        zc = __shfl(z, 16 + n, 32);
    }

    if (half == 0) {
        carr[(size_t)chunk * BS + b * S_DIM + s] = zc;
    }
}

// ---------------------------------------------------------------------------
// Kernel 2: scan the NCH chunk carries per column, seeded with z0.
// Converts carr[i] from "chunk-local final" to "incoming carry of chunk i".
// ---------------------------------------------------------------------------
__global__ void s4_pass2(const float* __restrict__ z0,
                         const float* __restrict__ log_neg_a,
                         const float* __restrict__ log_dt,
                         float* __restrict__ carr) {
    int idx = blockIdx.x * blockDim.x + threadIdx.x;   // over B*S
    if (idx >= BS) return;
    int s = idx & (S_DIM - 1);
    float ar = -expf(clampf_(log_neg_a[s], -8.0f, 4.0f));
    float dt = expf(clampf_(log_dt[s], -8.0f, 1.0f));
    float x  = dt * ar;
    float aT = expf((float)TG * x);                    // a_bar^TG
    float zc = z0[idx];
#pragma unroll
    for (int i = 0; i < NCH; ++i) {
        float lf = carr[(size_t)i * BS + idx];
        carr[(size_t)i * BS + idx] = zc;
        zc = fmaf(aT, zc, lf);
    }
}

// ---------------------------------------------------------------------------
// Kernel 3: fix-up pass: out[t] += a^{(t%TG)+1} * carry_in[chunk(t)]
// ---------------------------------------------------------------------------
__global__ void s4_pass3(const float* __restrict__ log_neg_a,
                         const float* __restrict__ log_dt,
                         const float* __restrict__ carr,
                         float* __restrict__ out) {
    size_t idx = (size_t)blockIdx.x * blockDim.x + threadIdx.x;  // over T*B*S
    int r = (int)(idx & (size_t)(BS - 1));   // b*S + s
    int t = (int)(idx >> 12);                // BS = 2^12
    int s = r & (S_DIM - 1);
    int m = t & (TG - 1);
    int chunk = t >> 8;                      // TG = 2^8
    float ar = -expf(clampf_(log_neg_a[s], -8.0f, 4.0f));
    float dt = expf(clampf_(log_dt[s], -8.0f, 1.0f));
    float x  = dt * ar;
    float w  = expf((float)(m + 1) * x);     // a_bar^(m+1)
    out[idx] = fmaf(w, carr[(size_t)chunk * BS + r], out[idx]);
}

// ---------------------------------------------------------------------------
extern "C" void kernel_launch(void* const* d_in, const int* in_sizes, int n_in,
                              void* d_out, int out_size, void* d_ws, size_t ws_size,
                              hipStream_t stream) {
    const float* inp  = (const float*)d_in[0];   // [T,B,D]
    const float* z0   = (const float*)d_in[1];   // [B,S]
    const float* lna  = (const float*)d_in[2];   // [S]
    const float* bmat = (const float*)d_in[3];   // [S,D]
    const float* ldt  = (const float*)d_in[4];   // [S]
    float* out = (float*)d_out;                  // [T,B,S]
    float* ws  = (float*)d_ws;

    // 0) parameter prep
    s4_prep<<<dim3(S_DIM), dim3(D_DIM), 0, stream>>>(lna, bmat, ldt, ws);

    // 1) per-chunk WMMA GEMM + local scan (4096 independent waves)
    s4_pass1<<<dim3(NCH, B_DIM, S_DIM / 16), dim3(32), 0, stream>>>(
        inp, ws, out, ws + WS_CARR);

    // 2) chunk-carry scan seeded with z0
    s4_pass2<<<dim3(BS / 256), dim3(256), 0, stream>>>(z0, lna, ldt, ws + WS_CARR);

    // 3) elementwise fix-up
    s4_pass3<<<dim3((unsigned)(((size_t)T_DIM * BS) / 256)), dim3(256), 0, stream>>>(
        lna, ldt, ws + WS_CARR, out);
}